// IDWT_1D_2439541424211
// MI455X (gfx1250) — compile-verified
//
#include <hip/hip_runtime.h>

// IDWT_1D (db4) as banded transposed-convolution, expressed as per-tile
// 16x16 WMMA f32 accumulations on gfx1250 (wave32).
//
//   out[row, p] = sum_k L[row, i0+k]*BL[k][p-p0] + H[row, i0+k]*BH[k][p-p0]
//   i0 = p0/2 - 2, K = 12, B fragments are tile-invariant filter taps.

typedef float v2f __attribute__((ext_vector_type(2)));
typedef float v8f __attribute__((ext_vector_type(8)));

#define N_ROWS 512      // 8*64
#define L_IN   4096
#define L_OUT  8192
#define CTILES (L_OUT / 16)   // 512
#define RTILES (N_ROWS / 16)  // 32
#define NTILES (CTILES * RTILES)

// Filter tap d (0..7) recovered from an interior row of the synthesis matrix:
// M[8, 20-d] == filter[d] for the db4 banded matrix built by the reference.
__device__ __forceinline__ float band_coef(const float* __restrict__ M, int d) {
  return (d >= 0 && d < 8) ? M[(size_t)8 * L_OUT + (20 - d)] : 0.0f;
}

__global__ __launch_bounds__(256) void idwt1d_wmma_kernel(
    const float* __restrict__ Lp, const float* __restrict__ Hp,
    const float* __restrict__ ML, const float* __restrict__ MH,
    float* __restrict__ out) {
  const int lane = threadIdx.x & 31;
  const int n = lane & 15;   // N column (B/C/D) and M row (A) for this lane
  const int h = lane >> 4;   // half-wave selector (K pairs / M+8)

  // Tile-invariant B fragments: B[k][j] = filter[2k - j], zero outside taps.
  // VGPR e of K-block kb holds K = 4*kb + 2*h + e, N = n.
  v2f bL[3], bH[3];
#pragma unroll
  for (int kb = 0; kb < 3; ++kb) {
    const int k0 = 4 * kb + 2 * h;
    bL[kb][0] = band_coef(ML, 2 * k0 - n);
    bL[kb][1] = band_coef(ML, 2 * (k0 + 1) - n);
    bH[kb][0] = band_coef(MH, 2 * k0 - n);
    bH[kb][1] = band_coef(MH, 2 * (k0 + 1) - n);
  }

  const int wavesPerBlock = blockDim.x >> 5;
  const int gw = blockIdx.x * wavesPerBlock + (threadIdx.x >> 5);
  const int nw = gridDim.x * wavesPerBlock;

  for (int tile = gw; tile < NTILES; tile += nw) {
    const int ct = tile & (CTILES - 1);
    const int rt = tile / CTILES;
    const int p0 = ct * 16;
    const int i0 = (p0 >> 1) - 2;          // first contributing input column
    const int ib = i0 + 2 * h;             // this lane's first K column
    const float* lrow = Lp + (size_t)(rt * 16 + n) * L_IN;
    const float* hrow = Hp + (size_t)(rt * 16 + n) * L_IN;

    v2f aL[3], aH[3];
    if (ct > 0 && ct < CTILES - 1) {
      // Interior tile: all 12 K columns in range; 8B-aligned float2 loads.
#pragma unroll
      for (int kb = 0; kb < 3; ++kb) {
        aL[kb] = *(const v2f*)(lrow + ib + 4 * kb);
        aH[kb] = *(const v2f*)(hrow + ib + 4 * kb);
      }
    } else {
      // Edge tiles (wave-uniform branch): clamp + zero out-of-range columns,
      // matching the reference's matrix truncation. EXEC stays all-ones.
#pragma unroll
      for (int kb = 0; kb < 3; ++kb) {
#pragma unroll
        for (int e = 0; e < 2; ++e) {
          const int i = ib + 4 * kb + e;
          const bool v = (i >= 0) && (i < L_IN);
          const int ic = v ? i : 0;
          aL[kb][e] = v ? lrow[ic] : 0.0f;
          aH[kb][e] = v ? hrow[ic] : 0.0f;
        }
      }
    }

    v8f c = {};
#pragma unroll
    for (int kb = 0; kb < 3; ++kb) {
      c = __builtin_amdgcn_wmma_f32_16x16x4_f32(
          /*neg_a=*/false, aL[kb], /*neg_b=*/false, bL[kb],
          /*c_mod=*/(short)0, c, /*reuse_a=*/false, /*reuse_b=*/false);
      c = __builtin_amdgcn_wmma_f32_16x16x4_f32(
          false, aH[kb], false, bH[kb], (short)0, c, false, false);
    }

    // D layout: VGPR r -> row M = r + 8*h, col N = n.
    float* orow = out + (size_t)(rt * 16 + h * 8) * L_OUT + p0 + n;
#pragma unroll
    for (int r = 0; r < 8; ++r) {
      orow[(size_t)r * L_OUT] = c[r];
    }
  }
}

extern "C" void kernel_launch(void* const* d_in, const int* in_sizes, int n_in,
                              void* d_out, int out_size, void* d_ws, size_t ws_size,
                              hipStream_t stream) {
  const float* L  = (const float*)d_in[0];
  const float* H  = (const float*)d_in[1];
  const float* ML = (const float*)d_in[2];
  const float* MH = (const float*)d_in[3];
  float* out = (float*)d_out;
  (void)in_sizes; (void)n_in; (void)out_size; (void)d_ws; (void)ws_size;

  // 16384 tiles; 8 waves per 256-thread block -> one tile per wave.
  dim3 grid(NTILES / 8), block(256);
  hipLaunchKernelGGL(idwt1d_wmma_kernel, grid, block, 0, stream, L, H, ML, MH, out);
}